// LightweightConv_7885559955944
// MI455X (gfx1250) — compile-verified
//
#include <hip/hip_runtime.h>

typedef __attribute__((ext_vector_type(2))) float v2f;
typedef __attribute__((ext_vector_type(8))) float v8f;

namespace {
constexpr int T_   = 4096;
constexpr int B_   = 8;
constexpr int C_   = 1024;
constexpr int H_   = 16;
constexpr int K_   = 31;
constexpr int RAD  = 15;              // pad_l = K/2
constexpr int CBLK = 16;              // channel tile = WMMA N
constexpr int NFRAG = 12;             // 12 * K=4 taps cover the 46-wide band
constexpr int TILES = 16;             // 16 time-tiles (256 steps) per wave
constexpr int STRIDE_T = B_ * C_;     // 8192 floats between consecutive t
constexpr int NCHUNK = (T_ / 16) / TILES;          // 16
constexpr int NCB    = C_ / CBLK;                  // 64
constexpr int WAVES  = B_ * NCB * NCHUNK;          // 8192
constexpr int WPB    = 4;                          // waves per block
}

// Load one B fragment half-pair. xrow = uniform pointer at absolute row `tstart`,
// srow = 4*s (compile-time const after unroll), voff = kbase*STRIDE + chan
// (per-lane loop-invariant), ta = absolute time of the .x element (guard only).
template <bool GUARD>
__device__ __forceinline__ v2f load_frag(const float* __restrict__ xrow,
                                         int srow, int voff, int ta) {
  v2f bv;
  if constexpr (GUARD) {
    bv.x = ((unsigned)ta       < (unsigned)T_) ? xrow[(long)srow * STRIDE_T + voff]            : 0.0f;
    bv.y = ((unsigned)(ta + 1) < (unsigned)T_) ? xrow[(long)srow * STRIDE_T + voff + STRIDE_T] : 0.0f;
  } else {
    bv.x = xrow[(long)srow * STRIDE_T + voff];
    bv.y = xrow[(long)srow * STRIDE_T + voff + STRIDE_T];
  }
  return bv;
}

template <bool GUARD>
__device__ __forceinline__ void chunk_loop(const float* __restrict__ x,
                                           float* __restrict__ out,
                                           const v2f* a, int t0_0,
                                           int voff, int svoff, int kbase) {
  int tstart = t0_0 - RAD;                               // band tap tau=0
  const float* __restrict__ xrow = x + (long)tstart * STRIDE_T;
  float* __restrict__ orow = out + (long)t0_0 * STRIDE_T;

  // Prime the 12-fragment sliding window (48 input rows in registers)
  v2f bf[NFRAG];
#pragma unroll
  for (int s = 0; s < NFRAG; ++s)
    bf[s] = load_frag<GUARD>(xrow, 4 * s, voff, tstart + 4 * s + kbase);

#pragma unroll
  for (int tile = 0; tile < TILES; ++tile) {
    if (tile > 0) {
      tstart += 16;
      xrow += (long)16 * STRIDE_T;
      orow += (long)16 * STRIDE_T;
      // shift-by-4: with full unroll these become register renames (no movs)
#pragma unroll
      for (int s = 0; s < NFRAG - 4; ++s) bf[s] = bf[s + 4];
#pragma unroll
      for (int s = NFRAG - 4; s < NFRAG; ++s)
        bf[s] = load_frag<GUARD>(xrow, 4 * s, voff, tstart + 4 * s + kbase);
    }

    if constexpr (!GUARD) {
      // stream-ahead prefetch; tstart+144 <= 3953 < T for all unguarded chunks
      __builtin_prefetch(xrow + (long)144 * STRIDE_T + voff, 0, 0);
    }

    // 16x16 output tile = 12 chained fp32 WMMAs over the banded weight matrix
    v8f acc = {};
#pragma unroll
    for (int s = 0; s < NFRAG; ++s) {
      acc = __builtin_amdgcn_wmma_f32_16x16x4_f32(
          /*neg_a=*/false, a[s], /*neg_b=*/false, bf[s],
          /*c_mod=*/(short)0, acc, /*reuse_a=*/false, /*reuse_b=*/false);
    }

    // store D: VGPR j -> row m = j + 8*half (folded into svoff), col n
#pragma unroll
    for (int j = 0; j < 8; ++j)
      __builtin_nontemporal_store(acc[j], &orow[(long)j * STRIDE_T + svoff]);
  }
}

__global__ __launch_bounds__(32 * WPB)
void lconv_wmma_f32(const float* __restrict__ x,
                    const float* __restrict__ w,
                    float* __restrict__ out) {
  const int lane  = threadIdx.x & 31;
  const int half  = lane >> 4;         // 0: K=0,1 / D rows 0-7 ; 1: K=2,3 / D rows 8-15
  const int mn    = lane & 15;         // A-row m == B/D-column n (per-lane index)
  const int kbase = half << 1;

  const int wid    = blockIdx.x * WPB + (threadIdx.x >> 5);
  const int tchunk = wid % NCHUNK;
  const int cb     = (wid / NCHUNK) % NCB;
  const int b      = wid / (NCHUNK * NCB);

  const int c0 = cb * CBLK;
  const float* __restrict__ wh = w + (c0 / (C_ / H_)) * K_;   // head = c0/64

  // A fragments: banded weight matrix A_s[m,kk] = w[4s+kk-m] (0 outside band)
  v2f a[NFRAG];
#pragma unroll
  for (int s = 0; s < NFRAG; ++s) {
    const int d0 = 4 * s + kbase - mn;
    v2f av;
    av.x = ((unsigned)d0       < (unsigned)K_) ? wh[d0]     : 0.0f;
    av.y = ((unsigned)(d0 + 1) < (unsigned)K_) ? wh[d0 + 1] : 0.0f;
    a[s] = av;
  }

  // Per-lane loop-invariant address components
  const int chan  = b * C_ + c0 + mn;             // channel column offset
  const int voff  = kbase * STRIDE_T + chan;      // B-fragment lane offset
  const int svoff = (half * 8) * STRIDE_T + chan; // D-store lane offset
  const int t0_0  = tchunk * (TILES * 16);

  if (tchunk == 0 || tchunk == NCHUNK - 1)
    chunk_loop<true>(x, out, a, t0_0, voff, svoff, kbase);
  else
    chunk_loop<false>(x, out, a, t0_0, voff, svoff, kbase);
}

extern "C" void kernel_launch(void* const* d_in, const int* in_sizes, int n_in,
                              void* d_out, int out_size, void* d_ws, size_t ws_size,
                              hipStream_t stream) {
  const float* x = (const float*)d_in[0];   // (T, B, C) fp32
  const float* w = (const float*)d_in[1];   // (H, 1, K) fp32
  float* out = (float*)d_out;               // (T, B, C) fp32

  const int nblocks = WAVES / WPB;          // 2048 blocks x 128 threads
  lconv_wmma_f32<<<nblocks, 32 * WPB, 0, stream>>>(x, w, out);
}